// VoxelSDFRenderer_29618094473683
// MI455X (gfx1250) — compile-verified
//
#include <hip/hip_runtime.h>

// VoxelSDF renderer for MI455X (gfx1250, wave32).
// Roofline: raw input = 207.6 MB dominates; 23.3 TB/s HBM -> ~9.3 us floor.
// Strategy: 1 wave per ray; shuffle-scan for visibility cumprod; fp32 WMMA
// (V_WMMA_F32_16X16X4_F32, exact precision) for the 128x24 feature matvec.
// fc-outer / kc-inner loop order keeps only ~70 VGPRs live (vs 64 accs alone
// in kc-outer), so loads issue as deep clauses and occupancy stays high;
// LDS transpose for coalesced NCHW output stores.

typedef float v2f __attribute__((ext_vector_type(2)));
typedef float v8f __attribute__((ext_vector_type(8)));

#define NS      24
#define RAW_CH  132
#define OUT_CH  135      // 3 rgb + 128 feat + 3 xyz + 1 mask
#define NPIX    16384    // 128*128
#define RAYS_PER_BLOCK 8

__device__ __forceinline__ float wred32(float v) {
#pragma unroll
    for (int off = 16; off; off >>= 1) v += __shfl_xor(v, off, 32);
    return v;  // full sum present in all lanes
}

__global__ __launch_bounds__(256)
void voxel_sdf_render(const float* __restrict__ rays,
                      const float* __restrict__ raw,
                      const float* __restrict__ sbeta,
                      float* __restrict__ out)
{
    __shared__ float lds[OUT_CH * 9];   // pad 8 -> 9 to kill bank conflicts

    const int lane = threadIdx.x & 31;
    const int wv   = threadIdx.x >> 5;                 // wave id in block = ray slot
    const int ray  = blockIdx.x * RAYS_PER_BLOCK + wv; // one wave per ray
    const int rbase = ray * (NS * RAW_CH);

    // ---- ray parameters (wave-uniform loads) ----
    const float* rp = rays + ray * 11;
    const float ox = rp[0], oy = rp[1], oz = rp[2];
    const float dx = rp[3], dy = rp[4], dz = rp[5];
    const float nr = rp[6], fr = rp[7];
    const float dn   = sqrtf(dx * dx + dy * dy + dz * dz);
    const float beta = sbeta[0];

    // ---- Phase A: per-sample alpha, visibility scan, weights (lane = sample) ----
    const int   sc   = (lane < NS) ? lane : (NS - 1);   // clamp for idle lanes
    const float t    = (float)sc * (1.0f / NS);         // t_vals[s] = s/24
    const float z    = nr + (fr - nr) * t;              // z_vals
    const float dist = (sc < NS - 1) ? (fr - nr) * (1.0f / NS) * dn
                                     : 1.0e10f * dn;

    // channels 0..3 (rgb + sdf) in one 16B-aligned b128 load
    const float4 head = *(const float4*)(raw + rbase + sc * RAW_CH);
    const float sdf = head.w;
    // sigmoid(-sdf/beta)/beta = 1 / (beta * (1 + exp(sdf/beta)))
    const float sigma = 1.0f / (beta * (1.0f + expf(sdf / beta)));
    const float alpha = 1.0f - expf(-sigma * dist);
    float trans = (lane < NS) ? (1.0f - alpha + 1e-10f) : 1.0f;

    // inclusive product scan over lanes, then shift for exclusive (visibility)
    float p = trans;
#pragma unroll
    for (int off = 1; off < 32; off <<= 1) {
        float q = __shfl_up(p, off, 32);
        if (lane >= off) p *= q;
    }
    float vis = __shfl_up(p, 1, 32);
    if (lane == 0) vis = 1.0f;

    float w = (lane < NS) ? alpha * vis : 0.0f;

    // w_last = 1 - sum(w[0..22])
    const float hsum = wred32((lane < NS - 1) ? w : 0.0f);
    const float w23  = 1.0f - hsum;
    if (lane == NS - 1) w = w23;

    // ---- rgb_map / xyz via wave reductions (lane = sample) ----
    const float rgbm0 = -1.0f + 2.0f * wred32(w / (1.0f + expf(-head.x)));
    const float rgbm1 = -1.0f + 2.0f * wred32(w / (1.0f + expf(-head.y)));
    const float rgbm2 = -1.0f + 2.0f * wred32(w / (1.0f + expf(-head.z)));
    const float xm = wred32(w * (ox + dx * z));
    const float ym = wred32(w * (oy + dy * z));
    const float zm = wred32(w * (oz + dz * z));

    if (lane == 0) {
        lds[0 * 9 + wv]   = rgbm0;
        lds[1 * 9 + wv]   = rgbm1;
        lds[2 * 9 + wv]   = rgbm2;
        lds[131 * 9 + wv] = xm;
        lds[132 * 9 + wv] = ym;
        lds[133 * 9 + wv] = zm;
        lds[134 * 9 + wv] = w23;
    }

    // ---- Phase B: feature_map = W (1x24) x F (24x128) via fp32 WMMA ----
    // A (16x4 f32): every row = weights chunk (M redundant).
    //   VGPR0: lanes0-15 K=0, lanes16-31 K=2 ; VGPR1: K=1 / K=3.
    // B (4x16 f32): VGPRv: lanes0-15 K=v, lanes16-31 K=v+2; N = lane&15.
    // D (16x16 f32): every lane's vgpr0 holds out[fc*16 + (lane&15)].
    const int hl   = lane >> 4;      // half-wave select
    const int nidx = lane & 15;      // N index (feature within chunk)

    // Precompute all A chunks up front (one ds_bpermute clause)
    v2f amat[6];
#pragma unroll
    for (int kc = 0; kc < 6; ++kc) {
        amat[kc].x = __shfl(w, kc * 4 + (hl ? 2 : 0), 32);
        amat[kc].y = __shfl(w, kc * 4 + (hl ? 3 : 1), 32);
    }

    const float* bp0 = raw + rbase + (hl ? 2 : 0) * RAW_CH + 4 + nidx;
    const float* bp1 = raw + rbase + (hl ? 3 : 1) * RAW_CH + 4 + nidx;

    float fval[8];   // per-lane feature results, stored after the MMA region
#pragma unroll
    for (int fc = 0; fc < 8; ++fc) {
        // 12-load clause for this feature chunk (disjoint columns per fc:
        // no duplicated traffic from fc-outer ordering)
        v2f b[6];
#pragma unroll
        for (int kc = 0; kc < 6; ++kc) {
            b[kc].x = bp0[kc * 4 * RAW_CH + fc * 16];
            b[kc].y = bp1[kc * 4 * RAW_CH + fc * 16];
        }
        // two partial accumulators -> two independent WMMA dependency chains
        v8f a0 = (v8f){0.f,0.f,0.f,0.f,0.f,0.f,0.f,0.f};
        v8f a1 = (v8f){0.f,0.f,0.f,0.f,0.f,0.f,0.f,0.f};
#pragma unroll
        for (int kc = 0; kc < 6; kc += 2) {
            a0 = __builtin_amdgcn_wmma_f32_16x16x4_f32(
                false, amat[kc],     false, b[kc],     (short)0, a0, false, false);
            a1 = __builtin_amdgcn_wmma_f32_16x16x4_f32(
                false, amat[kc + 1], false, b[kc + 1], (short)0, a1, false, false);
        }
        fval[fc] = a0[0] + a1[0];
    }

#pragma unroll
    for (int fc = 0; fc < 8; ++fc) {
        if (lane < 16)
            lds[(3 + fc * 16 + lane) * 9 + wv] = fval[fc];
    }

    __syncthreads();

    // ---- coalesced NCHW output: out[c*16384 + ray] ----
    const int obase = blockIdx.x * RAYS_PER_BLOCK;
    for (int i = threadIdx.x; i < OUT_CH * RAYS_PER_BLOCK; i += 256) {
        const int c = i >> 3;
        const int r = i & 7;
        out[c * NPIX + obase + r] = lds[c * 9 + r];
    }
}

extern "C" void kernel_launch(void* const* d_in, const int* in_sizes, int n_in,
                              void* d_out, int out_size, void* d_ws, size_t ws_size,
                              hipStream_t stream) {
    const float* rays = (const float*)d_in[0];   // (1,128,128,11)
    const float* raw  = (const float*)d_in[1];   // (1,128,128,24,132)
    const float* beta = (const float*)d_in[2];   // (1,)
    float* out = (float*)d_out;                  // (1,135,128,128)
    voxel_sdf_render<<<NPIX / RAYS_PER_BLOCK, 256, 0, stream>>>(rays, raw, beta, out);
}